// E3nnInteraction_75471165325414
// MI455X (gfx1250) — compile-verified
//
#include <hip/hip_runtime.h>
#include <hip/hip_bf16.h>
#include <math.h>

typedef float v2f __attribute__((ext_vector_type(2)));
typedef float v8f __attribute__((ext_vector_type(8)));

#define N_NODES   10000
#define N_EDGES   160000
#define MULCH     128      // "MUL" channels
#define RB        8        // R_BASIS
#define HID       64       // HIDDEN
#define WN        512      // W_NUMEL = 4*MUL

#define INV_SQRT128 0.08838834764831845f
#define INV_SQRT8   0.35355339059327373f
#define INV_8       0.125f
#define INV_SQRT3   0.5773502691896258f
#define SILU_2MOM   1.679177f
#define OUT_SCALE   0.00625f   // 1/(sqrt(256)*AVG_NUM_NEIGHBORS) = 1/160

__device__ __forceinline__ float silu_act(float v) {
    // SILU_2MOM * v * sigmoid(v): v_exp_f32 + v_add + v_rcp_f32 + v_mul
    // (exp and rcp are TRANS32 ops that co-execute with VALU/WMMA)
    const float e = __expf(-v);
    return SILU_2MOM * v * __builtin_amdgcn_rcpf(1.0f + e);
}

__device__ __forceinline__ v8f wmma4(v2f a, v2f b, v8f c) {
    return __builtin_amdgcn_wmma_f32_16x16x4_f32(false, a, false, b,
                                                 (short)0, c, false, false);
}

__device__ __forceinline__ void atomAddF32(float* p, float v) {
    unsafeAtomicAdd(p, v);   // native global_atomic_add_f32
}

// ---------------------------------------------------------------------------
// Kernel 1: up-projection, specialized per block BLK.
//   BLK 0: x0 @ W_up0 ; BLK 1..3: x1_{m=BLK-1} @ W_up1  (stride-3 A reads)
// ux[n][512]:  [0:128)=ux0, [128 + m*128 + u] = ux1[n][u][m]
// grid 625, block 128 (4 waves; each wave a 16x32 strip = 2 fragments).
// ---------------------------------------------------------------------------
template<int BLK>
__global__ __launch_bounds__(128)
void up_linear_kernel(const float* __restrict__ node_feats,
                      const float* __restrict__ Wm,
                      float* __restrict__ ux) {
    const int mtile = blockIdx.x;
    const int wave  = threadIdx.x >> 5;
    const int lane  = threadIdx.x & 31;
    const int g = lane >> 4, r = lane & 15;
    const int row = mtile * 16 + r;
    const int c0  = wave * 32 + r;          // col of fragment 0; frag1 = c0+16

    const float* arow = node_feats + (size_t)row * 512 + (BLK == 0 ? 0 : 128);
    v8f acc0 = {}, acc1 = {};
#pragma unroll 4
    for (int kk = 0; kk < 128; kk += 4) {
        const int k0 = kk + 2 * g;
        v2f a, b0, b1;
        if (BLK == 0) {
            a.x = arow[k0];
            a.y = arow[k0 + 1];
        } else {
            a.x = arow[(size_t)k0 * 3 + (BLK - 1)];
            a.y = arow[(size_t)(k0 + 1) * 3 + (BLK - 1)];
        }
        const float* wr = Wm + (size_t)k0 * 128;
        b0.x = wr[c0];        b0.y = wr[128 + c0];
        b1.x = wr[c0 + 16];   b1.y = wr[128 + c0 + 16];
        acc0 = wmma4(a, b0, acc0);
        acc1 = wmma4(a, b1, acc1);
    }
    float* dst = ux + (size_t)(mtile * 16) * 512 + BLK * 128;
#pragma unroll
    for (int i = 0; i < 8; ++i) {
        const int mr = i + 8 * g;
        dst[(size_t)mr * 512 + c0]      = acc0[i] * INV_SQRT128;
        dst[(size_t)mr * 512 + c0 + 16] = acc1[i] * INV_SQRT128;
    }
}

// ---------------------------------------------------------------------------
// Kernel 2: fused edge MLP (WMMA) + tensor product + atomic scatter.
// Each wave handles 16 edges. grid 2500, block 128 (4 waves).
// ---------------------------------------------------------------------------
__global__ __launch_bounds__(128)
void edge_fused_kernel(const float* __restrict__ edge_attrs,
                       const float* __restrict__ edge_feats,
                       const int*   __restrict__ edge_index,
                       const float* __restrict__ W_mlp1,
                       const float* __restrict__ W_mlp2,
                       const float* __restrict__ W_mlp3,
                       const float* __restrict__ W_mlp4,
                       const float* __restrict__ ux,
                       float*       __restrict__ msg) {
    __shared__ float lds_g[2][4][16 * 65];   // ping-pong activation buffers, padded
    __shared__ int   s_snd[4][16];
    __shared__ int   s_rcv[4][16];
    __shared__ float s_att[4][16][4];

    const int w    = threadIdx.x >> 5;
    const int lane = threadIdx.x & 31;
    const int g = lane >> 4, r = lane & 15;
    const int tile = blockIdx.x * 4 + w;     // 0..9999
    const int E0   = tile * 16;

    if (lane < 16) {
        const int e = E0 + lane;
        s_snd[w][lane] = edge_index[e];
        s_rcv[w][lane] = edge_index[N_EDGES + e];
#pragma unroll
        for (int q = 0; q < 4; ++q) s_att[w][lane][q] = edge_attrs[(size_t)e * 4 + q];
    }

    float* bufA = lds_g[0][w];
    float* bufB = lds_g[1][w];

    // ----- layer 1: (16x8) @ (8x64) -----
    {
        v8f c[4] = {{}, {}, {}, {}};
#pragma unroll
        for (int kk = 0; kk < RB; kk += 4) {
            const int k0 = kk + 2 * g;
            v2f a;
            a.x = edge_feats[(size_t)(E0 + r) * RB + k0];
            a.y = edge_feats[(size_t)(E0 + r) * RB + k0 + 1];
            const float* wr = W_mlp1 + (size_t)k0 * HID + r;
#pragma unroll
            for (int nt = 0; nt < 4; ++nt) {
                v2f b;
                b.x = wr[nt * 16];
                b.y = wr[HID + nt * 16];
                c[nt] = wmma4(a, b, c[nt]);
            }
        }
#pragma unroll
        for (int nt = 0; nt < 4; ++nt)
#pragma unroll
            for (int i = 0; i < 8; ++i)
                bufA[(i + 8 * g) * 65 + nt * 16 + r] = silu_act(c[nt][i] * INV_SQRT8);
    }
    __syncthreads();

    // ----- layers 2 and 3: (16x64) @ (64x64), ping-pong through LDS -----
#pragma unroll
    for (int layer = 0; layer < 2; ++layer) {
        const float* W   = (layer == 0) ? W_mlp2 : W_mlp3;
        const float* src = (layer == 0) ? bufA : bufB;
        float*       dst = (layer == 0) ? bufB : bufA;
        v8f c[4] = {{}, {}, {}, {}};
#pragma unroll 4
        for (int kk = 0; kk < HID; kk += 4) {
            const int k0 = kk + 2 * g;
            v2f a;
            a.x = src[r * 65 + k0];
            a.y = src[r * 65 + k0 + 1];
            const float* wr = W + (size_t)k0 * HID + r;
#pragma unroll
            for (int nt = 0; nt < 4; ++nt) {
                v2f b;
                b.x = wr[nt * 16];
                b.y = wr[HID + nt * 16];
                c[nt] = wmma4(a, b, c[nt]);
            }
        }
        __syncthreads();
#pragma unroll
        for (int nt = 0; nt < 4; ++nt)
#pragma unroll
            for (int i = 0; i < 8; ++i)
                dst[(i + 8 * g) * 65 + nt * 16 + r] = silu_act(c[nt][i] * INV_8);
        __syncthreads();
    }

    // ----- layer 4 + tensor product + scatter -----
    // For each 16-channel group cg: compute the w1..w4 tiles (4 fragments);
    // then each lane owns channel u = cg*16 + r for 8 edges (rows i + 8g).
    for (int cg = 0; cg < 8; ++cg) {
        v8f d[4] = {{}, {}, {}, {}};
#pragma unroll 4
        for (int kk = 0; kk < HID; kk += 4) {
            const int k0 = kk + 2 * g;
            v2f a;
            a.x = bufA[r * 65 + k0];
            a.y = bufA[r * 65 + k0 + 1];
            const float* wr = W_mlp4 + (size_t)k0 * WN + cg * 16 + r;
#pragma unroll
            for (int j = 0; j < 4; ++j) {
                v2f b;
                b.x = wr[j * MULCH];
                b.y = wr[WN + j * MULCH];
                d[j] = wmma4(a, b, d[j]);
            }
        }
        const int u = cg * 16 + r;
#pragma unroll
        for (int i = 0; i < 8; ++i) {
            const int el  = i + 8 * g;
            const int snd = s_snd[w][el];
            const int rcv = s_rcv[w][el];
            const float y0  = s_att[w][el][0];
            const float y1x = s_att[w][el][1];
            const float y1y = s_att[w][el][2];
            const float y1z = s_att[w][el][3];
            const float w1 = d[0][i] * INV_8;
            const float w2 = d[1][i] * INV_8;
            const float w3 = d[2][i] * INV_8;
            const float w4 = d[3][i] * INV_8;
            const float* uxr = ux + (size_t)snd * 512;
            const float s0  = uxr[u];
            const float s1x = uxr[128 + u];
            const float s1y = uxr[256 + u];
            const float s1z = uxr[384 + u];
            float* mr = msg + (size_t)rcv * 1024;
            atomAddF32(mr + u, w1 * s0 * y0);                       // p1
            const float dot = s1x * y1x + s1y * y1y + s1z * y1z;
            atomAddF32(mr + 128 + u, w4 * dot * INV_SQRT3);         // p4
            const float w2s0 = w2 * s0;
            atomAddF32(mr + 256 + u, w2s0 * y1x);                   // p2
            atomAddF32(mr + 512 + u, w2s0 * y1y);
            atomAddF32(mr + 768 + u, w2s0 * y1z);
            const float w3y0 = w3 * y0;
            atomAddF32(mr + 256 + 128 + u, w3y0 * s1x);             // p3
            atomAddF32(mr + 512 + 128 + u, w3y0 * s1y);
            atomAddF32(mr + 768 + 128 + u, w3y0 * s1z);
        }
    }
}

// ---------------------------------------------------------------------------
// Kernel 3: output projection, specialized per block BLK.
//   BLK 0: msg0 @ W_out0 -> out[:, 0:128)
//   BLK m+1: msg1_m @ W_out1 -> out[:, 128 + v*3 + m]
// grid 625, block 128 (4 waves; each wave 16x32).
// ---------------------------------------------------------------------------
template<int BLK>
__global__ __launch_bounds__(128)
void out_linear_kernel(const float* __restrict__ msg,
                       const float* __restrict__ Wm,
                       float* __restrict__ out) {
    const int mtile = blockIdx.x;
    const int wave  = threadIdx.x >> 5;
    const int lane  = threadIdx.x & 31;
    const int g = lane >> 4, r = lane & 15;
    const int row = mtile * 16 + r;
    const int c0  = wave * 32 + r;
    const size_t abase = (BLK == 0) ? 0 : (size_t)(256 + (BLK - 1) * 256);

    const float* arow = msg + (size_t)row * 1024 + abase;
    v8f acc0 = {}, acc1 = {};
#pragma unroll 4
    for (int kk = 0; kk < 256; kk += 4) {
        const int k0 = kk + 2 * g;
        v2f a, b0, b1;
        a.x = arow[k0];
        a.y = arow[k0 + 1];
        const float* wr = Wm + (size_t)k0 * 128;
        b0.x = wr[c0];        b0.y = wr[128 + c0];
        b1.x = wr[c0 + 16];   b1.y = wr[128 + c0 + 16];
        acc0 = wmma4(a, b0, acc0);
        acc1 = wmma4(a, b1, acc1);
    }
#pragma unroll
    for (int i = 0; i < 8; ++i) {
        const int n = mtile * 16 + i + 8 * g;
        const float v0 = acc0[i] * OUT_SCALE;
        const float v1 = acc1[i] * OUT_SCALE;
        if (BLK == 0) {
            out[(size_t)n * 512 + c0]      = v0;
            out[(size_t)n * 512 + c0 + 16] = v1;
        } else {
            out[(size_t)n * 512 + 128 + (size_t)c0 * 3 + (BLK - 1)]        = v0;
            out[(size_t)n * 512 + 128 + (size_t)(c0 + 16) * 3 + (BLK - 1)] = v1;
        }
    }
}

// ---------------------------------------------------------------------------
extern "C" void kernel_launch(void* const* d_in, const int* in_sizes, int n_in,
                              void* d_out, int out_size, void* d_ws, size_t ws_size,
                              hipStream_t stream) {
    const float* node_feats = (const float*)d_in[0];
    const float* edge_attrs = (const float*)d_in[1];
    const float* edge_feats = (const float*)d_in[2];
    const int*   edge_index = (const int*)  d_in[3];
    const float* W_up0  = (const float*)d_in[4];
    const float* W_up1  = (const float*)d_in[5];
    const float* W_mlp1 = (const float*)d_in[6];
    const float* W_mlp2 = (const float*)d_in[7];
    const float* W_mlp3 = (const float*)d_in[8];
    const float* W_mlp4 = (const float*)d_in[9];
    const float* W_out0 = (const float*)d_in[10];
    const float* W_out1 = (const float*)d_in[11];
    float* out = (float*)d_out;

    float* ux  = (float*)d_ws;                     // 10000*512 floats
    float* msg = ux + (size_t)N_NODES * 512;       // 10000*1024 floats

    hipMemsetAsync(msg, 0, (size_t)N_NODES * 1024 * sizeof(float), stream);

    up_linear_kernel<0><<<N_NODES / 16, 128, 0, stream>>>(node_feats, W_up0, ux);
    up_linear_kernel<1><<<N_NODES / 16, 128, 0, stream>>>(node_feats, W_up1, ux);
    up_linear_kernel<2><<<N_NODES / 16, 128, 0, stream>>>(node_feats, W_up1, ux);
    up_linear_kernel<3><<<N_NODES / 16, 128, 0, stream>>>(node_feats, W_up1, ux);

    edge_fused_kernel<<<N_EDGES / 64, 128, 0, stream>>>(
        edge_attrs, edge_feats, edge_index,
        W_mlp1, W_mlp2, W_mlp3, W_mlp4, ux, msg);

    out_linear_kernel<0><<<N_NODES / 16, 128, 0, stream>>>(msg, W_out0, out);
    out_linear_kernel<1><<<N_NODES / 16, 128, 0, stream>>>(msg, W_out1, out);
    out_linear_kernel<2><<<N_NODES / 16, 128, 0, stream>>>(msg, W_out1, out);
    out_linear_kernel<3><<<N_NODES / 16, 128, 0, stream>>>(msg, W_out1, out);
}